// QuantisedMSELoss_27728308863147
// MI455X (gfx1250) — compile-verified
//
#include <hip/hip_runtime.h>

typedef float v4f __attribute__((ext_vector_type(4)));
typedef float v2f __attribute__((ext_vector_type(2)));
typedef float v8f __attribute__((ext_vector_type(8)));

#define NBINS   10
#define NQ      20          // 10 bin sums + 10 bin counts
#define BLK     256
#define NBLOCKS 1024
#define HSTRIDE 23          // padded per-thread hist stride (gcd(23,64)=1 -> bank-conflict-free)

// ---------------------------------------------------------------------------
// Kernel 1: streaming pass. Per-thread-private LDS histograms, ds_add_f32
// atomics (single-writer -> deterministic), nontemporal b128 global loads.
// Each block emits one padded row of 32 floats into d_ws (cols 20..31 = 0).
// ---------------------------------------------------------------------------
__global__ __launch_bounds__(BLK) void qmse_partial_kernel(
    const float* __restrict__ target,
    const float* __restrict__ output,
    const float* __restrict__ edges,
    float* __restrict__ ws,
    long n)
{
    __shared__ float hist[BLK * HSTRIDE];   // per-thread private histograms
    __shared__ float part[240];             // combine stage scratch
    __shared__ float finals[NQ];

    const int tid = threadIdx.x;
    float* __restrict__ h = &hist[tid * HSTRIDE];

    // inner quantile edges 1..9 (uniform -> scalar loads)
    float e[9];
#pragma unroll
    for (int j = 0; j < 9; ++j) e[j] = edges[j + 1];

#pragma unroll
    for (int q = 0; q < NQ; ++q) h[q] = 0.0f;
    __syncthreads();

    const long n4 = n >> 2;
    const v4f* __restrict__ t4 = (const v4f*)target;
    const v4f* __restrict__ o4 = (const v4f*)output;
    const long stride = (long)gridDim.x * BLK;

    for (long i = (long)blockIdx.x * BLK + tid; i < n4; i += stride) {
        v4f tv = __builtin_nontemporal_load(t4 + i);   // global_load_b128 th:NT
        v4f ov = __builtin_nontemporal_load(o4 + i);
#pragma unroll
        for (int c = 0; c < 4; ++c) {
            float t = tv[c], o = ov[c];
            bool valid = (t == t);                     // NaN targets masked
            float d = t - o;
            float sq = valid ? d * d : 0.0f;
            // side='left' searchsorted, clipped: count of inner edges strictly below t
            int b = 0;
#pragma unroll
            for (int j = 0; j < 9; ++j) b += (t > e[j]) ? 1 : 0;
            // single-instruction no-return LDS float adds (private copy)
            __hip_atomic_fetch_add(&h[b], sq,
                                   __ATOMIC_RELAXED, __HIP_MEMORY_SCOPE_WORKGROUP);
            __hip_atomic_fetch_add(&h[NBINS + b], valid ? 1.0f : 0.0f,
                                   __ATOMIC_RELAXED, __HIP_MEMORY_SCOPE_WORKGROUP);
        }
    }
    // scalar tail (n not divisible by 4): one thread, into its own private hist
    if (blockIdx.x == 0 && tid == 0) {
        for (long k = n4 << 2; k < n; ++k) {
            float t = target[k], o = output[k];
            bool valid = (t == t);
            float d = t - o;
            float sq = valid ? d * d : 0.0f;
            int b = 0;
#pragma unroll
            for (int j = 0; j < 9; ++j) b += (t > e[j]) ? 1 : 0;
            h[b] += sq;
            h[NBINS + b] += valid ? 1.0f : 0.0f;
        }
    }
    __syncthreads();

    // Deterministic combine of 256 private hists -> 20 block values.
    // 240 threads: quantity q = tid%20, copy-group g = tid/20 (12 groups x 22 copies).
    if (tid < 240) {
        int q = tid % NQ;
        int g = tid / NQ;
        int c0 = g * 22;
        int c1 = c0 + 22; if (c1 > BLK) c1 = BLK;
        float s = 0.0f;
        for (int c = c0; c < c1; ++c) s += hist[c * HSTRIDE + q];
        part[tid] = s;
    }
    __syncthreads();
    if (tid < NQ) {
        float s = 0.0f;
#pragma unroll
        for (int g = 0; g < 12; ++g) s += part[tid + NQ * g];   // fixed order
        finals[tid] = s;
    }
    __syncthreads();
    if (tid < 32) ws[(long)blockIdx.x * 32 + tid] = (tid < NQ) ? finals[tid] : 0.0f;
}

// ---------------------------------------------------------------------------
// Kernel 2: reduce the 1024x32 partial matrix with V_WMMA_F32_16X16X4_F32
// (A = ones -> column sums, exact f32, fixed order), then compute the loss.
// 8 waves: waves 0-3 -> columns 0..15, waves 4-7 -> columns 16..31; each wave
// handles a quarter of the rows; cross-wave combine in fixed order via LDS.
// ---------------------------------------------------------------------------
__global__ __launch_bounds__(BLK) void qmse_final_kernel(
    const float* __restrict__ ws,
    float* __restrict__ out,
    int nrows)
{
    __shared__ float wsum[8 * 16];
    __shared__ float finals[32];

    const int tid  = threadIdx.x;
    const int lane = tid & 31;
    const int wave = tid >> 5;
    const int colgrp  = wave >> 2;           // 0 or 1
    const int quarter = wave & 3;            // row quarter
    const int col  = colgrp * 16 + (lane & 15);
    const int rows_per = nrows >> 2;         // rows per wave
    const int rk   = (lane & 16) ? 2 : 0;    // K rows 0/1 in lanes 0-15, 2/3 in 16-31

    v2f a; a[0] = 1.0f; a[1] = 1.0f;         // ones A-matrix (16x4)
    v8f acc = {};

    const int rbase = quarter * rows_per;
    for (int i = 0; i < rows_per; i += 4) {
        int r = rbase + i + rk;
        v2f b;
        b[0] = ws[(long)r * 32 + col];
        b[1] = ws[(long)(r + 1) * 32 + col];
        // D = ones(16x4) x B(4x16) + C  -> every D row = column sums of B
        acc = __builtin_amdgcn_wmma_f32_16x16x4_f32(
            false, a, false, b, (short)0, acc, false, false);
    }
    // acc[0] holds the column sum for `col` in every lane; emit one copy per wave
    if (lane < 16) wsum[wave * 16 + lane] = acc[0];
    __syncthreads();

    if (tid < 32) {
        int cg = tid >> 4;
        float s = 0.0f;
#pragma unroll
        for (int w = 0; w < 4; ++w) s += wsum[(cg * 4 + w) * 16 + (tid & 15)];
        finals[tid] = s;                     // fixed order -> deterministic
    }
    __syncthreads();

    if (tid == 0) {
        float ts = 0.0f, tc = 0.0f, qs = 0.0f, ne = 0.0f;
#pragma unroll
        for (int b = 0; b < NBINS; ++b) {
            float s = finals[b];
            float c = finals[NBINS + b];
            ts += s;
            tc += c;
            if (c > 0.0f) { qs += s / (c * c); ne += 1.0f; }
        }
        float mse  = ts / fmaxf(tc, 1.0f);
        float qmse = (ne > 0.0f) ? (qs / ne) : 0.0f;
        out[0] = mse + 1.0f * qmse;          // ALPHA = 1
    }
}

// ---------------------------------------------------------------------------
extern "C" void kernel_launch(void* const* d_in, const int* in_sizes, int n_in,
                              void* d_out, int out_size, void* d_ws, size_t ws_size,
                              hipStream_t stream) {
    (void)n_in; (void)out_size; (void)ws_size;
    const float* target = (const float*)d_in[0];
    const float* output = (const float*)d_in[1];
    const float* edges  = (const float*)d_in[2];
    float* ws  = (float*)d_ws;               // needs NBLOCKS*32*4 = 128 KB
    float* out = (float*)d_out;
    long n = (long)in_sizes[0];

    qmse_partial_kernel<<<NBLOCKS, BLK, 0, stream>>>(target, output, edges, ws, n);
    qmse_final_kernel<<<1, BLK, 0, stream>>>(ws, out, NBLOCKS);
}